// GINBasedNet_55594056679592
// MI455X (gfx1250) — compile-verified
//
#include <hip/hip_runtime.h>

typedef __attribute__((ext_vector_type(2))) float v2f;
typedef __attribute__((ext_vector_type(8))) float v8f;

#define NN 50000
#define EE 800000
#define DIM 128
#define NCLS 40
#define CP 48   // classes padded to a multiple of 16

// ---------------------------------------------------------------- utility
__global__ void zero_kernel(float* __restrict__ p, long n) {
    long i = (long)blockIdx.x * blockDim.x + threadIdx.x;
    long stride = (long)gridDim.x * blockDim.x;
    for (; i < n; i += stride) p[i] = 0.0f;
}

__global__ void deg_kernel(const int* __restrict__ dst, float* __restrict__ deg) {
    int e = blockIdx.x * blockDim.x + threadIdx.x;
    if (e < EE) unsafeAtomicAdd(&deg[dst[e]], 1.0f);
}

__global__ void deginv_kernel(float* __restrict__ deg) {
    int i = blockIdx.x * blockDim.x + threadIdx.x;
    if (i < NN) {
        float d = deg[i];
        deg[i] = (d > 0.0f) ? (1.0f / d) : 0.0f;
    }
}

// fold BN constants: scale = g*rsqrt(rv+eps), shift = (b-rm)*scale + beta
__global__ void consts_kernel(const float* b0, const float* g0, const float* be0,
                              const float* rm0, const float* rv0,
                              const float* b1, const float* g1, const float* be1,
                              const float* rm1, const float* rv1,
                              const float* b2, const float* g2, const float* be2,
                              const float* rm2, const float* rv2,
                              float* __restrict__ cbuf) {
    int t = threadIdx.x;
    const float eps = 1e-5f;
    if (t < DIM) {
        float s = g0[t] * rsqrtf(rv0[t] + eps);
        cbuf[0 * DIM + t] = s;
        cbuf[1 * DIM + t] = (b0[t] - rm0[t]) * s + be0[t];
        s = g1[t] * rsqrtf(rv1[t] + eps);
        cbuf[2 * DIM + t] = s;
        cbuf[3 * DIM + t] = (b1[t] - rm1[t]) * s + be1[t];
    }
    if (t < NCLS) {
        float s = g2[t] * rsqrtf(rv2[t] + eps);
        cbuf[4 * DIM + t]      = s;
        cbuf[4 * DIM + 64 + t] = (b2[t] - rm2[t]) * s + be2[t];
    }
}

// ----------------------------------------------------- edge scatter (mean agg)
// one wave per edge: 32 lanes x float4 = 128 floats of the source row
__global__ void scatter_kernel(const float* __restrict__ x,
                               const int* __restrict__ src,
                               const int* __restrict__ dst,
                               const float* __restrict__ mask,
                               float* __restrict__ agg) {
    long idx = (long)blockIdx.x * blockDim.x + threadIdx.x;
    long total = (long)EE * (DIM / 4);
    if (idx >= total) return;
    int e = (int)(idx >> 5);        // DIM/4 == 32 float4 per row
    int c = (int)((idx & 31) << 2); // starting column
    int s = src[e];
    int d = dst[e];
    float m = mask[e];
    const float4 v = *(const float4*)(x + (long)s * DIM + c);
    float* ap = agg + (long)d * DIM + c;
    unsafeAtomicAdd(ap + 0, v.x * m);
    unsafeAtomicAdd(ap + 1, v.y * m);
    unsafeAtomicAdd(ap + 2, v.z * m);
    unsafeAtomicAdd(ap + 3, v.w * m);
}

// -------------------------------------------------- GIN layer GEMM (128->128)
// out = relu( ((x + agg*deginv) @ W) * scale + shift )
// W held in LDS in k-pair-interleaved layout: sW[p*DIM + c] = {W[2p][c], W[2p+1][c]}
// so a B fragment is a single aligned ds_load_b64 into the WMMA source pair.
__global__ void __launch_bounds__(256)
gin_gemm_kernel(const float* __restrict__ x, const float* __restrict__ agg,
                const float* __restrict__ dinv, const float* __restrict__ W,
                const float* __restrict__ scale, const float* __restrict__ shift,
                float* __restrict__ out) {
    __shared__ v2f sW[(DIM / 2) * DIM]; // 64 KB
    const int tid = threadIdx.x;
    for (int i = tid; i < (DIM / 2) * DIM; i += 256) {
        int p = i / DIM, c = i % DIM;
        v2f w;
        w.x = W[(2 * p) * DIM + c];
        w.y = W[(2 * p + 1) * DIM + c];
        sW[i] = w;
    }
    __syncthreads();

    const int wave = tid >> 5;
    const int lane = tid & 31;
    const int m    = lane & 15;   // row within tile (A) / col within tile (B)
    const int prow = lane >> 4;   // which k-pair of the k-step: 0 or 1
    const int kk   = prow << 1;   // K offset: 0 or 2
    const int row0 = blockIdx.x * 128 + wave * 16;

    const int  r     = row0 + m;
    const bool valid = r < NN;
    const int  rc    = valid ? r : 0;
    const float di   = valid ? dinv[rc] : 0.0f;
    const float* xr  = x   + (long)rc * DIM;
    const float* ar  = agg + (long)rc * DIM;

    v8f acc[8] = {};
    // software-pipelined A fragment (one k-step ahead)
    float2 xv = *(const float2*)(xr + kk);
    float2 av = *(const float2*)(ar + kk);
#pragma unroll 4
    for (int k0 = 0; k0 < DIM; k0 += 4) {
        // batch all 8 B fragments for this k-step: single-wait ds_load_b64 group
        const v2f* bb = sW + ((k0 >> 1) + prow) * DIM + m;
        v2f bf[8];
#pragma unroll
        for (int nt = 0; nt < 8; ++nt) bf[nt] = bb[nt * 16];

        v2f a;
        a.x = valid ? (xv.x + av.x * di) : 0.0f;
        a.y = valid ? (xv.y + av.y * di) : 0.0f;

        if (k0 + 4 < DIM) {
            xv = *(const float2*)(xr + k0 + 4 + kk);
            av = *(const float2*)(ar + k0 + 4 + kk);
        }
#pragma unroll
        for (int nt = 0; nt < 8; ++nt)
            acc[nt] = __builtin_amdgcn_wmma_f32_16x16x4_f32(
                false, a, false, bf[nt], (short)0, acc[nt], false, false);
    }

    // epilogue: BN + relu.  C/D layout: VGPR j, lanes0-15 -> M=j, lanes16-31 -> M=8+j
    const int roff = prow << 3;
#pragma unroll
    for (int nt = 0; nt < 8; ++nt) {
        const int col = nt * 16 + m;
        const float s  = scale[col];
        const float sh = shift[col];
#pragma unroll
        for (int j = 0; j < 8; ++j) {
            int rr = row0 + roff + j;
            if (rr < NN) {
                float v = acc[nt][j] * s + sh;
                out[(long)rr * DIM + col] = v > 0.0f ? v : 0.0f;
            }
        }
    }
}

// ------------------------------------- fused final GIN layer + prediction head
// h3 = relu(BN((h2 + agg*dinv) @ w2)),  pred = h2 @ wp + bp
// out = 0.5 * (pred + h3)
__global__ void __launch_bounds__(256)
gin_final_kernel(const float* __restrict__ h2, const float* __restrict__ agg,
                 const float* __restrict__ dinv,
                 const float* __restrict__ w2, const float* __restrict__ wp,
                 const float* __restrict__ scale2, const float* __restrict__ shift2,
                 const float* __restrict__ bp, float* __restrict__ out) {
    // k-pair interleaved, cols padded to 48
    __shared__ v2f sW2[(DIM / 2) * CP]; // 24 KB
    __shared__ v2f sWP[(DIM / 2) * CP]; // 24 KB
    const int tid = threadIdx.x;
    for (int i = tid; i < (DIM / 2) * CP; i += 256) {
        int p = i / CP, c = i % CP;
        v2f a = {0.0f, 0.0f}, b = {0.0f, 0.0f};
        if (c < NCLS) {
            a.x = w2[(2 * p) * NCLS + c];
            a.y = w2[(2 * p + 1) * NCLS + c];
            b.x = wp[(2 * p) * NCLS + c];
            b.y = wp[(2 * p + 1) * NCLS + c];
        }
        sW2[i] = a;
        sWP[i] = b;
    }
    __syncthreads();

    const int wave = tid >> 5;
    const int lane = tid & 31;
    const int m    = lane & 15;
    const int prow = lane >> 4;
    const int kk   = prow << 1;
    const int row0 = blockIdx.x * 128 + wave * 16;

    const int  r     = row0 + m;
    const bool valid = r < NN;
    const int  rc    = valid ? r : 0;
    const float di   = valid ? dinv[rc] : 0.0f;
    const float* xr  = h2  + (long)rc * DIM;
    const float* ar  = agg + (long)rc * DIM;

    v8f acc3[3] = {};
    v8f accp[3] = {};
    float2 xv = *(const float2*)(xr + kk);
    float2 av = *(const float2*)(ar + kk);
#pragma unroll 4
    for (int k0 = 0; k0 < DIM; k0 += 4) {
        const int pidx = ((k0 >> 1) + prow) * CP + m;
        v2f bf2[3], bfp[3];
#pragma unroll
        for (int nt = 0; nt < 3; ++nt) {
            bf2[nt] = sW2[pidx + nt * 16];
            bfp[nt] = sWP[pidx + nt * 16];
        }

        v2f a3, ap;
        a3.x = valid ? (xv.x + av.x * di) : 0.0f;
        a3.y = valid ? (xv.y + av.y * di) : 0.0f;
        ap.x = valid ? xv.x : 0.0f;
        ap.y = valid ? xv.y : 0.0f;

        if (k0 + 4 < DIM) {
            xv = *(const float2*)(xr + k0 + 4 + kk);
            av = *(const float2*)(ar + k0 + 4 + kk);
        }
#pragma unroll
        for (int nt = 0; nt < 3; ++nt) {
            acc3[nt] = __builtin_amdgcn_wmma_f32_16x16x4_f32(
                false, a3, false, bf2[nt], (short)0, acc3[nt], false, false);
            accp[nt] = __builtin_amdgcn_wmma_f32_16x16x4_f32(
                false, ap, false, bfp[nt], (short)0, accp[nt], false, false);
        }
    }

    const int roff = prow << 3;
#pragma unroll
    for (int nt = 0; nt < 3; ++nt) {
        const int col = nt * 16 + m;
        if (col >= NCLS) continue;
        const float s  = scale2[col];
        const float sh = shift2[col];
        const float bb = bp[col];
#pragma unroll
        for (int j = 0; j < 8; ++j) {
            int rr = row0 + roff + j;
            if (rr < NN) {
                float h3 = acc3[nt][j] * s + sh;
                h3 = h3 > 0.0f ? h3 : 0.0f;
                out[(long)rr * NCLS + col] = 0.5f * (accp[nt][j] + bb + h3);
            }
        }
    }
}

// ---------------------------------------------------------------- launcher
extern "C" void kernel_launch(void* const* d_in, const int* in_sizes, int n_in,
                              void* d_out, int out_size, void* d_ws, size_t ws_size,
                              hipStream_t stream) {
    const float* h        = (const float*)d_in[0];
    const int*   edge_src = (const int*)d_in[1];
    const int*   edge_dst = (const int*)d_in[2];
    const float* mask     = (const float*)d_in[3];
    const float* w0  = (const float*)d_in[4];
    const float* b0  = (const float*)d_in[5];
    const float* g0  = (const float*)d_in[6];
    const float* be0 = (const float*)d_in[7];
    const float* rm0 = (const float*)d_in[8];
    const float* rv0 = (const float*)d_in[9];
    const float* w1  = (const float*)d_in[10];
    const float* b1  = (const float*)d_in[11];
    const float* g1  = (const float*)d_in[12];
    const float* be1 = (const float*)d_in[13];
    const float* rm1 = (const float*)d_in[14];
    const float* rv1 = (const float*)d_in[15];
    const float* w2  = (const float*)d_in[16];
    const float* b2  = (const float*)d_in[17];
    const float* g2  = (const float*)d_in[18];
    const float* be2 = (const float*)d_in[19];
    const float* rm2 = (const float*)d_in[20];
    const float* rv2 = (const float*)d_in[21];
    const float* wp  = (const float*)d_in[22];
    const float* bp  = (const float*)d_in[23];
    float* out = (float*)d_out;

    // workspace carving (floats)
    float* W = (float*)d_ws;
    float* dinv = W;                        // NN (padded)
    float* agg  = dinv + 50048;             // NN*DIM
    float* h1   = agg  + (long)NN * DIM;    // NN*DIM
    float* h2   = h1   + (long)NN * DIM;    // NN*DIM
    float* cbuf = h2   + (long)NN * DIM;    // 4*128 + 2*64 consts
    float* sc0 = cbuf + 0 * DIM;
    float* sh0 = cbuf + 1 * DIM;
    float* sc1 = cbuf + 2 * DIM;
    float* sh1 = cbuf + 3 * DIM;
    float* sc2 = cbuf + 4 * DIM;
    float* sh2 = cbuf + 4 * DIM + 64;

    const long aggN = (long)NN * DIM;
    const int ZB = 256;
    const int zeroBlocksAgg = (int)((aggN + ZB * 8 - 1) / (ZB * 8));
    const int gemmBlocks = (NN + 127) / 128;

    // degree -> deg_inv
    zero_kernel<<<(NN + ZB - 1) / ZB, ZB, 0, stream>>>(dinv, NN);
    deg_kernel<<<(EE + ZB - 1) / ZB, ZB, 0, stream>>>(edge_dst, dinv);
    deginv_kernel<<<(NN + ZB - 1) / ZB, ZB, 0, stream>>>(dinv);

    // folded BN constants
    consts_kernel<<<1, 128, 0, stream>>>(b0, g0, be0, rm0, rv0,
                                         b1, g1, be1, rm1, rv1,
                                         b2, g2, be2, rm2, rv2, cbuf);

    const long scatThreads = (long)EE * (DIM / 4);
    const int scatBlocks = (int)((scatThreads + ZB - 1) / ZB);

    // layer 0: h -> h1
    zero_kernel<<<zeroBlocksAgg, ZB, 0, stream>>>(agg, aggN);
    scatter_kernel<<<scatBlocks, ZB, 0, stream>>>(h, edge_src, edge_dst, mask, agg);
    gin_gemm_kernel<<<gemmBlocks, 256, 0, stream>>>(h, agg, dinv, w0, sc0, sh0, h1);

    // layer 1: h1 -> h2
    zero_kernel<<<zeroBlocksAgg, ZB, 0, stream>>>(agg, aggN);
    scatter_kernel<<<scatBlocks, ZB, 0, stream>>>(h1, edge_src, edge_dst, mask, agg);
    gin_gemm_kernel<<<gemmBlocks, 256, 0, stream>>>(h1, agg, dinv, w1, sc1, sh1, h2);

    // layer 2 + prediction head fused -> out
    zero_kernel<<<zeroBlocksAgg, ZB, 0, stream>>>(agg, aggN);
    scatter_kernel<<<scatBlocks, ZB, 0, stream>>>(h2, edge_src, edge_dst, mask, agg);
    gin_final_kernel<<<gemmBlocks, 256, 0, stream>>>(h2, agg, dinv, w2, wp,
                                                     sc2, sh2, bp, out);
}